// FourierSeries_84696755077155
// MI455X (gfx1250) — compile-verified
//
#include <hip/hip_runtime.h>

// FourierSeries on MI455X (gfx1250, wave32, WMMA).
// Pass 1: phase = x @ coef (fp32 WMMA 16x16x4), fused sin/cos -> d_ws.
// Pass 2: gain = cosP @ an^T + sinP @ bn^T + H*bias (fp32 WMMA), -> d_out.
// Workspace requirement: 2 * 4096 * 2048 * 4 B = 64 MB (fully rewritten each call).
//
// All LDS tiles use [row][k] layout with row stride 20 floats so every WMMA
// fragment (A and B) is a single aligned ds_load_b64 into an adjacent VGPR
// pair -- no register re-packing movs in the inner loop.

typedef __attribute__((ext_vector_type(2))) float v2f;
typedef __attribute__((ext_vector_type(8))) float v8f;

namespace {
constexpr int kB = 4096;  // batch
constexpr int kI = 2048;  // input features (gain output dim)
constexpr int kH = 2048;  // harmonics (phase dim)

constexpr int BM = 128;   // block tile M
constexpr int BN = 128;   // block tile N
constexpr int BK = 16;    // K chunk (4 WMMA k-steps of 4)
constexpr int LDA = BK + 4;  // 20 floats: float4-aligned stores, bank-spread b64 reads
}  // namespace

// ---------------------------------------------------------------------------
// Kernel 1: phase GEMM + sincos epilogue.
//   phase[m,n] = sum_k x[m,k] * coef[k,n];  cosP/sinP[m,n] = cos/sin(phase)
// coef tile is stored transposed ([n][k]) in LDS so B fragments are b64 loads.
// ---------------------------------------------------------------------------
__global__ __launch_bounds__(256)
void phase_trig_kernel(const float* __restrict__ x,
                       const float* __restrict__ coef,
                       float* __restrict__ cosP,
                       float* __restrict__ sinP) {
  __shared__ float As[BM * LDA];    // A tile, [m][k]
  __shared__ float BsT[BN * LDA];   // B tile, [n][k] (transposed coef chunk)

  const int tid   = threadIdx.x;
  const int lane  = tid & 31;
  const int wave  = tid >> 5;       // 0..7
  const int wm    = wave & 3;       // 4 waves along M
  const int wn    = wave >> 2;      // 2 waves along N
  const int lm    = lane & 15;      // M (A) / N (B,C) index within fragment
  const int lk    = (lane >> 4) << 1;  // K pair base: 0 or 2
  const int lhalf = (lane >> 4) << 3;  // C-row offset: 0 or 8

  const int m0 = blockIdx.y * BM;
  const int n0 = blockIdx.x * BN;

  v8f acc[2][4] = {};

  for (int k0 = 0; k0 < kI; k0 += BK) {
    __syncthreads();  // previous chunk's compute done before overwriting LDS

#pragma unroll
    for (int i = 0; i < 2; ++i) {
      const int idx = i * 256 + tid;

      // A tile: 128 rows x 16 k, one float4 per unit, b128 in and out.
      const int r = idx >> 2;            // 0..127 (m)
      const int c = (idx & 3) << 2;      // 0,4,8,12 (k)
      const float4 va = *(const float4*)(x + (size_t)(m0 + r) * kI + k0 + c);
      *(float4*)&As[r * LDA + c] = va;

      // B tile: gather a k-column of 4 (coalesced across lanes in n),
      // store transposed as one b128 into [n][k].
      const int nb = idx & 127;          // 0..127 (n)
      const int cb = (idx >> 7) << 2;    // 0,4 (i=0) / 8,12 (i=1)
      float4 vb;
      vb.x = coef[(size_t)(k0 + cb + 0) * kH + n0 + nb];
      vb.y = coef[(size_t)(k0 + cb + 1) * kH + n0 + nb];
      vb.z = coef[(size_t)(k0 + cb + 2) * kH + n0 + nb];
      vb.w = coef[(size_t)(k0 + cb + 3) * kH + n0 + nb];
      *(float4*)&BsT[nb * LDA + cb] = vb;
    }
    __syncthreads();

    if (k0 + BK < kI) {  // gfx1250 global_prefetch_b8 for next chunk
      __builtin_prefetch(x + (size_t)(m0 + (tid >> 2)) * kI + k0 + BK + ((tid & 3) << 2), 0, 0);
      __builtin_prefetch(coef + (size_t)(k0 + BK + (tid >> 4)) * kH + n0 + ((tid & 15) << 3), 0, 0);
    }

#pragma unroll
    for (int ks = 0; ks < BK; ks += 4) {
      v2f af[2], bf[4];
#pragma unroll
      for (int mt = 0; mt < 2; ++mt)
        af[mt] = *(const v2f*)&As[(wm * 32 + mt * 16 + lm) * LDA + ks + lk];
#pragma unroll
      for (int nt = 0; nt < 4; ++nt)
        bf[nt] = *(const v2f*)&BsT[(wn * 64 + nt * 16 + lm) * LDA + ks + lk];
#pragma unroll
      for (int mt = 0; mt < 2; ++mt)
#pragma unroll
        for (int nt = 0; nt < 4; ++nt)
          acc[mt][nt] = __builtin_amdgcn_wmma_f32_16x16x4_f32(
              false, af[mt], false, bf[nt], (short)0, acc[mt][nt], false, false);
    }
  }

  // Epilogue: sincos and store both trig planes.
#pragma unroll
  for (int mt = 0; mt < 2; ++mt) {
#pragma unroll
    for (int nt = 0; nt < 4; ++nt) {
      const int col  = n0 + wn * 64 + nt * 16 + lm;
      const int row0 = m0 + wm * 32 + mt * 16 + lhalf;
#pragma unroll
      for (int v = 0; v < 8; ++v) {
        const float p = acc[mt][nt][v];
        const size_t off = (size_t)(row0 + v) * kH + col;
        cosP[off] = __cosf(p);
        sinP[off] = __sinf(p);
      }
    }
  }
}

// ---------------------------------------------------------------------------
// Kernel 2: gain[m,i] = sum_h cosP[m,h]*an[i,h] + sinP[m,h]*bn[i,h] + H*bias[i]
// B matrices are an^T/bn^T; an is [I,H] row-major so tiles load contiguously
// along k and drop straight into an [n][k] LDS layout -> b64 fragment loads.
// ---------------------------------------------------------------------------
__global__ __launch_bounds__(256)
void gain_kernel(const float* __restrict__ cosP,
                 const float* __restrict__ sinP,
                 const float* __restrict__ an,
                 const float* __restrict__ bn,
                 const float* __restrict__ bias,
                 float* __restrict__ out) {
  __shared__ float Ac[BM * LDA];    // cos tile [m][k]
  __shared__ float Asn[BM * LDA];   // sin tile [m][k]
  __shared__ float Ban[BN * LDA];   // an tile  [n][k]  (an^T column tile)
  __shared__ float Bbn[BN * LDA];   // bn tile  [n][k]

  const int tid   = threadIdx.x;
  const int lane  = tid & 31;
  const int wave  = tid >> 5;
  const int wm    = wave & 3;
  const int wn    = wave >> 2;
  const int lm    = lane & 15;
  const int lk    = (lane >> 4) << 1;
  const int lhalf = (lane >> 4) << 3;

  const int m0 = blockIdx.y * BM;   // batch rows
  const int n0 = blockIdx.x * BN;   // output features (I)

  v8f acc[2][4] = {};

  for (int k0 = 0; k0 < kH; k0 += BK) {
    __syncthreads();
#pragma unroll
    for (int i = 0; i < 2; ++i) {
      const int idx = i * 256 + tid;
      const int r = idx >> 2;          // 0..127
      const int c = (idx & 3) << 2;    // 0,4,8,12 (k offset)
      const float4 vc = *(const float4*)(cosP + (size_t)(m0 + r) * kH + k0 + c);
      *(float4*)&Ac[r * LDA + c] = vc;
      const float4 vs = *(const float4*)(sinP + (size_t)(m0 + r) * kH + k0 + c);
      *(float4*)&Asn[r * LDA + c] = vs;
      const float4 va = *(const float4*)(an + (size_t)(n0 + r) * kH + k0 + c);
      *(float4*)&Ban[r * LDA + c] = va;
      const float4 vb = *(const float4*)(bn + (size_t)(n0 + r) * kH + k0 + c);
      *(float4*)&Bbn[r * LDA + c] = vb;
    }
    __syncthreads();

    if (k0 + BK < kH) {
      __builtin_prefetch(cosP + (size_t)(m0 + (tid >> 2)) * kH + k0 + BK, 0, 0);
      __builtin_prefetch(sinP + (size_t)(m0 + (tid >> 2)) * kH + k0 + BK, 0, 0);
      __builtin_prefetch(an + (size_t)(n0 + (tid >> 2)) * kH + k0 + BK, 0, 0);
      __builtin_prefetch(bn + (size_t)(n0 + (tid >> 2)) * kH + k0 + BK, 0, 0);
    }

#pragma unroll
    for (int ks = 0; ks < BK; ks += 4) {
      v2f afc[2], afs[2], bfa[4], bfb[4];
#pragma unroll
      for (int mt = 0; mt < 2; ++mt) {
        const int ro = (wm * 32 + mt * 16 + lm) * LDA + ks + lk;
        afc[mt] = *(const v2f*)&Ac[ro];
        afs[mt] = *(const v2f*)&Asn[ro];
      }
#pragma unroll
      for (int nt = 0; nt < 4; ++nt) {
        const int ro = (wn * 64 + nt * 16 + lm) * LDA + ks + lk;
        bfa[nt] = *(const v2f*)&Ban[ro];
        bfb[nt] = *(const v2f*)&Bbn[ro];
      }
#pragma unroll
      for (int mt = 0; mt < 2; ++mt)
#pragma unroll
        for (int nt = 0; nt < 4; ++nt) {
          acc[mt][nt] = __builtin_amdgcn_wmma_f32_16x16x4_f32(
              false, afc[mt], false, bfa[nt], (short)0, acc[mt][nt], false, false);
          acc[mt][nt] = __builtin_amdgcn_wmma_f32_16x16x4_f32(
              false, afs[mt], false, bfb[nt], (short)0, acc[mt][nt], false, false);
        }
    }
  }

  // Epilogue: + num_harmonics * bias[i]
#pragma unroll
  for (int nt = 0; nt < 4; ++nt) {
    const int col = n0 + wn * 64 + nt * 16 + lm;
    const float bv = (float)kH * bias[col];
#pragma unroll
    for (int mt = 0; mt < 2; ++mt) {
      const int row0 = m0 + wm * 32 + mt * 16 + lhalf;
#pragma unroll
      for (int v = 0; v < 8; ++v)
        out[(size_t)(row0 + v) * kI + col] = acc[mt][nt][v] + bv;
    }
  }
}

// ---------------------------------------------------------------------------
extern "C" void kernel_launch(void* const* d_in, const int* in_sizes, int n_in,
                              void* d_out, int out_size, void* d_ws, size_t ws_size,
                              hipStream_t stream) {
  (void)in_sizes; (void)n_in; (void)out_size; (void)ws_size;

  const float* x    = (const float*)d_in[0];  // [B, I]
  const float* coef = (const float*)d_in[1];  // [I, H]
  const float* an   = (const float*)d_in[2];  // [1, I, H]
  const float* bn   = (const float*)d_in[3];  // [1, I, H]
  const float* bias = (const float*)d_in[4];  // [1, I]
  float* out = (float*)d_out;                 // [B, I]

  float* cosP = (float*)d_ws;                 // [B, H]
  float* sinP = cosP + (size_t)kB * kH;       // [B, H]

  const dim3 blk(256);
  const dim3 g1(kH / BN, kB / BM);  // 16 x 32
  phase_trig_kernel<<<g1, blk, 0, stream>>>(x, coef, cosP, sinP);

  const dim3 g2(kI / BN, kB / BM);  // 16 x 32
  gain_kernel<<<g2, blk, 0, stream>>>(cosP, sinP, an, bn, bias, out);
}